// MMVP_Model_75376676044949
// MI455X (gfx1250) — compile-verified
//
#include <hip/hip_runtime.h>

// MMVP collapse: out[b,t] = (HornerT[b] @ last[b,t]) rows 0..31, normalized by row 32.
// HornerT: T = A_0; for k=0..8: T = T @ prev[b,k] + A_{k+1}; A_p = [feats[b,p]; ones; 0-pad]
// All f32 via V_WMMA_F32_16X16X4_F32 chained over K=64; sim matrices double-buffered in LDS.

typedef __attribute__((ext_vector_type(2))) float v2f;
typedef __attribute__((ext_vector_type(8))) float v8f;
typedef __attribute__((ext_vector_type(4))) float f4;

#define PREV_LEN 10
#define AFT_LEN  10
#define NB       256
#define NC       32
#define HW       64
#define TM       (PREV_LEN - 1 + AFT_LEN)   // 19
#define MROWS    48                          // 33 live rows (32 feat + 1 ones), zero-padded
#define TSTR     68                          // LDS row stride (floats): pads banks, keeps 16B align
#define EPSF     1e-6f

__global__ __launch_bounds__(128)
void mmvp_kernel(const float* __restrict__ feats,
                 const float* __restrict__ sim,
                 float* __restrict__ out)
{
    __shared__ float sT[MROWS * TSTR];     // running Horner matrix, 48 x 64 (stride 68)
    __shared__ float sM[2][HW * HW];       // double-buffered 64x64 sim matrix

    const int b    = blockIdx.x;
    const int tid  = threadIdx.x;
    const int wv   = tid >> 5;             // 4 waves: one 16-col output strip each
    const int lane = tid & 31;
    const int lo16 = lane & 15;
    const int hi   = lane >> 4;            // 0: K 0/1 + rows r / 1: K 2/3 + rows r+8
    const int cw   = wv * 16 + lo16;       // this lane's output column (0..63)

    const float* featsB = feats + (size_t)b * PREV_LEN * NC * HW;
    const float* simB   = sim   + (size_t)b * TM * HW * HW;

    // ---- stage m_0 into buffer 0 ----
    {
        const f4* s4 = (const f4*)simB;
        f4* d4 = (f4*)sM[0];
        for (int idx = tid; idx < (HW * HW) / 4; idx += 128)
            d4[idx] = s4[idx];
    }

    // ---- T = [feats[b,0] (32x64); ones (1x64); zeros (15x64)] ----
    for (int idx = tid; idx < MROWS * HW; idx += 128) {
        int r = idx >> 6, c = idx & 63;
        float v = 0.0f;
        if (r < NC)       v = featsB[r * HW + c];
        else if (r == NC) v = 1.0f;
        sT[r * TSTR + c] = v;
    }
    __syncthreads();

    for (int step = 0; step < TM; ++step) {
        const float* __restrict__ cur = sM[step & 1];

        // stage m_{step+1} into the other buffer; its ds-store wait lands at the
        // end-of-step barrier, so HBM latency overlaps the 48 WMMAs below.
        if (step + 1 < TM) {
            const f4* s4 = (const f4*)(simB + (size_t)(step + 1) * HW * HW);
            f4* d4 = (f4*)sM[(step + 1) & 1];
            for (int idx = tid; idx < (HW * HW) / 4; idx += 128)
                d4[idx] = s4[idx];
        }
        // prefetch m_{step+2} (global_prefetch_b8): 128B/thread covers 16KB
        if (step + 2 < TM)
            __builtin_prefetch(simB + (size_t)(step + 2) * HW * HW + tid * 32, 0, 0);

        const bool isPrev = (step < PREV_LEN - 1);

        // accumulators (C operand of WMMA): preload additive term A_{step+1}
        v8f a0 = {}, a1 = {}, a2 = {};
        if (isPrev) {
            const float* fa = featsB + (size_t)(step + 1) * NC * HW;
            #pragma unroll
            for (int r = 0; r < 8; ++r) {
                int r0 = r + hi * 8;
                a0[r] = fa[(r0)      * HW + cw];   // rows 0..15
                a1[r] = fa[(r0 + 16) * HW + cw];   // rows 16..31
            }
            a2[0] = (hi == 0) ? 1.0f : 0.0f;       // row 32 = ones; rows 33..47 stay 0
        }

        // K = 64 as 16 chained V_WMMA_F32_16X16X4_F32 per tile, 3 M-tiles.
        // cur & sT were made visible by the previous iteration's barrier.
        #pragma unroll
        for (int kk = 0; kk < 16; ++kk) {
            const int kb = kk * 4 + hi * 2;        // K-slot pair per half-wave
            v2f bf, fA0, fA1, fA2;
            bf.x  = cur[(kb)     * HW + cw];
            bf.y  = cur[(kb + 1) * HW + cw];
            fA0.x = sT[(lo16)      * TSTR + kb];
            fA0.y = sT[(lo16)      * TSTR + kb + 1];
            fA1.x = sT[(lo16 + 16) * TSTR + kb];
            fA1.y = sT[(lo16 + 16) * TSTR + kb + 1];
            fA2.x = sT[(lo16 + 32) * TSTR + kb];
            fA2.y = sT[(lo16 + 32) * TSTR + kb + 1];
            a0 = __builtin_amdgcn_wmma_f32_16x16x4_f32(false, fA0, false, bf, (short)0, a0, false, false);
            a1 = __builtin_amdgcn_wmma_f32_16x16x4_f32(false, fA1, false, bf, (short)0, a1, false, false);
            a2 = __builtin_amdgcn_wmma_f32_16x16x4_f32(false, fA2, false, bf, (short)0, a2, false, false);
        }
        __syncthreads();   // all sT/cur reads done; next-buffer stage flushed & visible

        if (isPrev) {
            // write new T back (D layout: vgpr r -> row tile*16 + r (+8 for hi lanes))
            #pragma unroll
            for (int r = 0; r < 8; ++r) {
                int r0 = r + hi * 8;
                sT[(r0)      * TSTR + cw] = a0[r];
                sT[(r0 + 16) * TSTR + cw] = a1[r];
                sT[(r0 + 32) * TSTR + cw] = a2[r];
            }
            __syncthreads();
        } else {
            // output step t: row 32 of T@last_t is the column-weight sum
            const int t = step - (PREV_LEN - 1);
            float wrow  = a2[0];                        // lanes 0..15 hold row 32
            float wsum  = __shfl(wrow, lo16, 32);       // broadcast across half-waves
            float denom = wsum + EPSF;
            float* ob = out + (size_t)(b * AFT_LEN + t) * NC * HW;
            #pragma unroll
            for (int r = 0; r < 8; ++r) {
                int r0 = r + hi * 8;
                ob[(r0)      * HW + cw] = a0[r] / denom;
                ob[(r0 + 16) * HW + cw] = a1[r] / denom;
            }
        }
    }
}

extern "C" void kernel_launch(void* const* d_in, const int* in_sizes, int n_in,
                              void* d_out, int out_size, void* d_ws, size_t ws_size,
                              hipStream_t stream) {
    const float* feats = (const float*)d_in[0];   // (256,10,32,8,8) f32
    const float* sim   = (const float*)d_in[1];   // (256,19,64,64)  f32
    float* out = (float*)d_out;                   // (2560,32,8,8)   f32
    mmvp_kernel<<<NB, 128, 0, stream>>>(feats, sim, out);
}